// MassTool_78640851190236
// MI455X (gfx1250) — compile-verified
//
#include <hip/hip_runtime.h>

#define Qn 16000
#define Tn 10000
#define Sn 1000
#define Dn 768
#define Kn 8

typedef float vf2 __attribute__((ext_vector_type(2)));
typedef float vf8 __attribute__((ext_vector_type(8)));
typedef unsigned int u32x4 __attribute__((ext_vector_type(4)));
typedef int i32x4 __attribute__((ext_vector_type(4)));
typedef int i32x8 __attribute__((ext_vector_type(8)));

// ---------------------------------------------------------------------------
// SpMM (COO): out[dst] += val * x[src], x is a virtual concat [xa (rowsA rows); xb]
// One block (192 threads) per edge; each thread handles one float4 (768 = 192*4).
// Edge-parallel fp32 atomics: feature matrices fit in the 192MB L2, so the
// global_atomic_add_f32 stream is L2-resident.
// ---------------------------------------------------------------------------
__global__ __launch_bounds__(192)
void spmm_kernel(const int* __restrict__ src, const int* __restrict__ dst,
                 const float* __restrict__ val,
                 const float* __restrict__ xa, int rowsA,
                 const float* __restrict__ xb,
                 float* __restrict__ out, int nEdges) {
    int e = blockIdx.x;
    if (e >= nEdges) return;
    int s = src[e];
    int d = dst[e];
    float v = val[e];
    const float* xr = (s < rowsA) ? (xa + (size_t)s * Dn)
                                  : (xb + (size_t)(s - rowsA) * Dn);
    float* orow = out + (size_t)d * Dn;
    int c4 = threadIdx.x * 4;
    float4 xv = *(const float4*)(xr + c4);
    atomicAdd(orow + c4 + 0, v * xv.x);
    atomicAdd(orow + c4 + 1, v * xv.y);
    atomicAdd(orow + c4 + 2, v * xv.z);
    atomicAdd(orow + c4 + 3, v * xv.w);
}

// ---------------------------------------------------------------------------
// acc = (x + h1 + h2) / 3, x read from virtual concat [xa ; xb] (split at splitElems)
// outbuf may alias h1 (same-index elementwise).
// ---------------------------------------------------------------------------
__global__ __launch_bounds__(256)
void combine3_kernel(const float* __restrict__ xa, long splitElems,
                     const float* __restrict__ xb,
                     const float* __restrict__ h1, const float* __restrict__ h2,
                     float* __restrict__ outbuf, long n) {
    long i = (long)blockIdx.x * blockDim.x + threadIdx.x;
    if (i >= n) return;
    float x = (i < splitElems) ? xa[i] : xb[i - splitElems];
    outbuf[i] = (x + h1[i] + h2[i]) * (1.0f / 3.0f);
}

// ---------------------------------------------------------------------------
// TDM: issue tensor_load_to_lds of a 64x64 fp32 tile (row stride N=768 floats)
// into LDS at byte offset lds_off. D# per CDNA5 ISA ch.8:
//   group0: count=1 | lds_addr[63:32] | global_addr[120:64] | type=2[127:126]
//   group1: data_size=4B, tensor_dim0/1=768, tile_dim0/1=64, dim0_stride=768
//   groups 2/3 (and trailing group): zero (2D tile)
// This toolchain exposes the 6-arg builtin:
//   (u32x4 g0, i32x8 g1, i32x4 g2, i32x4 g3, i32x8 g4, i32 cpol)
// ---------------------------------------------------------------------------
__device__ __forceinline__ void tdm_load_panel_64x64(const float* gsrc, unsigned lds_off) {
    unsigned long long ga = (unsigned long long)(uintptr_t)gsrc;
    u32x4 g0;
    g0[0] = 1u;                                                   // count=1 (valid user D#)
    g0[1] = lds_off;                                              // LDS byte address
    g0[2] = (unsigned)(ga & 0xFFFFFFFFu);                         // global addr [31:0]
    g0[3] = ((unsigned)(ga >> 32) & 0x01FFFFFFu) | 0x80000000u;   // addr [56:32] | type=2
    i32x8 g1;
    g1[0] = (int)(2u << 16);        // data_size=2 (4 bytes)
    g1[1] = (int)(768u << 16);      // tensor_dim0 = 768 (bits 79:48, lo16)
    g1[2] = (int)(768u << 16);      // tensor_dim0 hi=0 | tensor_dim1 lo16 = 768
    g1[3] = (int)(64u << 16);       // tensor_dim1 hi=0 | tile_dim0 = 64
    g1[4] = 64;                     // tile_dim1 = 64 | tile_dim2 = 0
    g1[5] = 768;                    // tensor_dim0_stride[31:0] = 768
    g1[6] = 0;                      // stride hi | tensor_dim1_stride lo
    g1[7] = 0;
    i32x4 gz4 = {0, 0, 0, 0};
    i32x8 gz8 = {0, 0, 0, 0, 0, 0, 0, 0};
    __builtin_amdgcn_tensor_load_to_lds(g0, g1, gz4, gz4, gz8, 0);
}

// ---------------------------------------------------------------------------
// GEMM C = A(MxK) * B(KxN), fp32, via V_WMMA_F32_16X16X4_F32.
// One block (8 waves) owns one 64-column strip of B and 8 consecutive 16-row
// M tiles of A. B is staged through LDS in 64x64 chunks by the Tensor Data
// Mover (wave 0 issues tensor_load_to_lds, double-buffered, TENSORcnt +
// workgroup barrier for publication). Each wave: 4 accumulators (16x64 strip),
// A fragment (global_load_b64) reused 4x, B fragments from ds_load.
// ---------------------------------------------------------------------------
__global__ __launch_bounds__(256)
void gemm_wmma_f32(const float* __restrict__ A, const float* __restrict__ B,
                   float* __restrict__ Cmat, int M, int N, int Kdim) {
    __shared__ float bpanel[2][64 * 64];

    const int lane = threadIdx.x & 31;
    const int wave = threadIdx.x >> 5;
    const int nStrips = N >> 6;                 // 12 strips of 64 columns
    const int bm = blockIdx.x / nStrips;        // M super-tile (8 x 16 rows)
    const int nt = blockIdx.x - bm * nStrips;
    const int mt = bm * 8 + wave;               // this wave's 16-row tile
    const int half = lane >> 4;                 // 0: lanes 0-15, 1: lanes 16-31
    const int l16  = lane & 15;
    const int row  = mt * 16 + l16;
    const int n0   = nt * 64;

    const float* arow = A + (size_t)row * Kdim;
    const int nChunks = Kdim >> 6;              // 12 chunks of K=64

    vf8 acc0 = {}, acc1 = {}, acc2 = {}, acc3 = {};

    // prologue: DMA chunk 0 into buffer 0
    if (wave == 0) {
        tdm_load_panel_64x64(B + (size_t)n0, (unsigned)(uintptr_t)&bpanel[0][0]);
        __builtin_amdgcn_s_wait_tensorcnt(0);
    }
    __syncthreads();

    for (int c = 0; c < nChunks; ++c) {
        const int cur = c & 1;
        if (wave == 0 && c + 1 < nChunks) {
            // previous-iteration barrier guarantees buffer cur^1 is free
            tdm_load_panel_64x64(B + (size_t)(c + 1) * 64 * N + n0,
                                 (unsigned)(uintptr_t)&bpanel[cur ^ 1][0]);
        }

        const float* bl = &bpanel[cur][0];
        #pragma unroll 4
        for (int kk = 0; kk < 64; kk += 4) {
            const int ka = c * 64 + kk + half * 2;
            vf2 af;
            af.x = arow[ka];
            af.y = arow[ka + 1];
            const float* bp0 = bl + (kk + half * 2) * 64 + l16;
            const float* bp1 = bp0 + 64;
            vf2 b0, b1, b2, b3;
            b0.x = bp0[0];  b0.y = bp1[0];
            b1.x = bp0[16]; b1.y = bp1[16];
            b2.x = bp0[32]; b2.y = bp1[32];
            b3.x = bp0[48]; b3.y = bp1[48];
            acc0 = __builtin_amdgcn_wmma_f32_16x16x4_f32(false, af, false, b0, (short)0, acc0, false, false);
            acc1 = __builtin_amdgcn_wmma_f32_16x16x4_f32(false, af, false, b1, (short)0, acc1, false, false);
            acc2 = __builtin_amdgcn_wmma_f32_16x16x4_f32(false, af, false, b2, (short)0, acc2, false, false);
            acc3 = __builtin_amdgcn_wmma_f32_16x16x4_f32(false, af, false, b3, (short)0, acc3, false, false);
        }

        if (c + 1 < nChunks) {
            if (wave == 0) __builtin_amdgcn_s_wait_tensorcnt(0);
            __syncthreads();   // publish next buffer; also fences reads of cur
        }
    }

    #pragma unroll
    for (int j = 0; j < 8; ++j) {
        int rr = mt * 16 + half * 8 + j;
        float* crow = Cmat + (size_t)rr * N + l16;
        crow[n0 + 0]  = acc0[j];
        crow[n0 + 16] = acc1[j];
        crow[n0 + 32] = acc2[j];
        crow[n0 + 48] = acc3[j];
    }
}

// ---------------------------------------------------------------------------
// Per-query attention (K=8) over two banks + final combine:
// out = 0.5*(q_tl + att_t) + 0.5*(q_sl + att_s). One wave32 per query,
// 24 columns per lane, shfl_xor reductions for the 768-length dots.
// ---------------------------------------------------------------------------
__global__ __launch_bounds__(256)
void attn_final_kernel(const float* __restrict__ q_tl, const float* __restrict__ q_sl,
                       const float* __restrict__ qw_t, const float* __restrict__ qw_s,
                       const int* __restrict__ nbr, float* __restrict__ out) {
    const int lane = threadIdx.x & 31;
    const int q = (blockIdx.x * blockDim.x + threadIdx.x) >> 5;
    if (q >= Qn) return;

    const int CPL = Dn / 32;   // 24 columns per lane
    float wt[CPL], wsv[CPL];
    #pragma unroll
    for (int j = 0; j < CPL; ++j) {
        int c = lane + j * 32;
        wt[j]  = qw_t[(size_t)q * Dn + c];
        wsv[j] = qw_s[(size_t)q * Dn + c];
    }

    int nb[Kn];
    #pragma unroll
    for (int k = 0; k < Kn; ++k) nb[k] = nbr[q * Kn + k];

    float st[Kn], ss[Kn];
    #pragma unroll
    for (int k = 0; k < Kn; ++k) {
        const float* kt = q_tl + (size_t)nb[k] * Dn;
        const float* ks = q_sl + (size_t)nb[k] * Dn;
        float at = 0.f, as = 0.f;
        #pragma unroll
        for (int j = 0; j < CPL; ++j) {
            int c = lane + j * 32;
            at += wt[j]  * kt[c];
            as += wsv[j] * ks[c];
        }
        #pragma unroll
        for (int off = 16; off > 0; off >>= 1) {
            at += __shfl_xor(at, off, 32);
            as += __shfl_xor(as, off, 32);
        }
        st[k] = at;
        ss[k] = as;
    }

    // softmax over K=8 (all lanes hold identical reduced values)
    float mt = st[0], ms = ss[0];
    #pragma unroll
    for (int k = 1; k < Kn; ++k) { mt = fmaxf(mt, st[k]); ms = fmaxf(ms, ss[k]); }
    float et[Kn], es[Kn];
    float sumt = 0.f, sums = 0.f;
    #pragma unroll
    for (int k = 0; k < Kn; ++k) {
        et[k] = __expf(st[k] - mt); sumt += et[k];
        es[k] = __expf(ss[k] - ms); sums += es[k];
    }
    float rt = 1.f / sumt, rs = 1.f / sums;

    float acc[CPL];
    #pragma unroll
    for (int j = 0; j < CPL; ++j) {
        int c = lane + j * 32;
        acc[j] = q_tl[(size_t)q * Dn + c] + q_sl[(size_t)q * Dn + c];
    }
    #pragma unroll
    for (int k = 0; k < Kn; ++k) {
        float wkt = et[k] * rt;
        float wks = es[k] * rs;
        const float* kt = q_tl + (size_t)nb[k] * Dn;
        const float* ks = q_sl + (size_t)nb[k] * Dn;
        #pragma unroll
        for (int j = 0; j < CPL; ++j) {
            int c = lane + j * 32;
            acc[j] += wkt * kt[c] + wks * ks[c];
        }
    }
    #pragma unroll
    for (int j = 0; j < CPL; ++j) {
        int c = lane + j * 32;
        out[(size_t)q * Dn + c] = 0.5f * acc[j];
    }
}

// ---------------------------------------------------------------------------
extern "C" void kernel_launch(void* const* d_in, const int* in_sizes, int n_in,
                              void* d_out, int out_size, void* d_ws, size_t ws_size,
                              hipStream_t stream) {
    const float* qf     = (const float*)d_in[0];
    const float* tf     = (const float*)d_in[1];
    const float* w_t    = (const float*)d_in[2];
    const float* w_s    = (const float*)d_in[3];
    const float* qt_val = (const float*)d_in[4];
    const float* qs_val = (const float*)d_in[5];
    const float* st_val = (const float*)d_in[6];
    const int*   qt_src = (const int*)d_in[7];
    const int*   qt_dst = (const int*)d_in[8];
    const int*   qs_src = (const int*)d_in[9];
    const int*   qs_dst = (const int*)d_in[10];
    const int*   st_src = (const int*)d_in[11];
    const int*   st_dst = (const int*)d_in[12];
    const int*   nbr    = (const int*)d_in[13];
    const int Eqt = in_sizes[4];
    const int Eqs = in_sizes[5];
    const int Est = in_sizes[6];

    const size_t NQT = (size_t)(Qn + Tn) * Dn;   // 19,968,000
    const size_t NQS = (size_t)(Qn + Sn) * Dn;   // 13,056,000
    const size_t NS  = (size_t)Sn * Dn;          //    768,000
    const size_t NQ  = (size_t)Qn * Dn;          // 12,288,000

    float* A  = (float*)d_ws;        // h1_qt -> h_qt (q_tl | t_tl)
    float* B  = A + NQT;             // h2_qt -> h1_qs -> h_qs (q_sl)
    float* C  = B + NQT;             // h2_qs -> qw_t
    float* SC = C + NQS;             // scenes
    float* E  = SC + NS;             // qw_s

    // --- LightGCN on (queries|tools) graph ---
    (void)hipMemsetAsync(A, 0, NQT * sizeof(float), stream);
    spmm_kernel<<<Eqt, 192, 0, stream>>>(qt_src, qt_dst, qt_val, qf, Qn, tf, A, Eqt);
    (void)hipMemsetAsync(B, 0, NQT * sizeof(float), stream);
    spmm_kernel<<<Eqt, 192, 0, stream>>>(qt_src, qt_dst, qt_val, A, Qn + Tn, A, B, Eqt);
    {
        long n = (long)NQT;
        combine3_kernel<<<(unsigned)((n + 255) / 256), 256, 0, stream>>>(
            qf, (long)NQ, tf, A, B, A, n);   // A := h_qt
    }

    // --- scenes = segment_sum over st graph of t_tl ---
    (void)hipMemsetAsync(SC, 0, NS * sizeof(float), stream);
    spmm_kernel<<<Est, 192, 0, stream>>>(st_src, st_dst, st_val, A + NQ, Tn, A + NQ, SC, Est);

    // --- LightGCN on (queries|scenes) graph ---
    (void)hipMemsetAsync(B, 0, NQS * sizeof(float), stream);
    spmm_kernel<<<Eqs, 192, 0, stream>>>(qs_src, qs_dst, qs_val, qf, Qn, SC, B, Eqs);
    (void)hipMemsetAsync(C, 0, NQS * sizeof(float), stream);
    spmm_kernel<<<Eqs, 192, 0, stream>>>(qs_src, qs_dst, qs_val, B, Qn + Sn, B, C, Eqs);
    {
        long n = (long)NQS;
        combine3_kernel<<<(unsigned)((n + 255) / 256), 256, 0, stream>>>(
            qf, (long)NQ, SC, B, C, B, n);   // B := h_qs
    }

    // --- qw_t = q_tl @ w_t ; qw_s = q_sl @ w_s  (WMMA fp32, TDM-staged B) ---
    {
        int blocks = (Qn / 16 / 8) * (Dn / 64);   // 125 * 12 = 1500
        gemm_wmma_f32<<<blocks, 256, 0, stream>>>(A, w_t, C, Qn, Dn, Dn); // qw_t -> C
        gemm_wmma_f32<<<blocks, 256, 0, stream>>>(B, w_s, E, Qn, Dn, Dn); // qw_s -> E
    }

    // --- attention + final combine ---
    {
        int blocks = (Qn * 32 + 255) / 256;  // one wave32 per query
        attn_final_kernel<<<blocks, 256, 0, stream>>>(A, B, C, E, nbr, (float*)d_out);
    }
}